// ElementalReadOut_80728205296199
// MI455X (gfx1250) — compile-verified
//
#include <hip/hip_runtime.h>
#include <hip/hip_bf16.h>

typedef __attribute__((ext_vector_type(2))) float v2f;
typedef __attribute__((ext_vector_type(8))) float v8f;

#define N_CRYSTALS 2048
#define D_FEA 64
#define N_HEAD 32

// ---------------------------------------------------------------------------
// f32 WMMA 16x16x4 wrapper: D = A(16x4) * B(4x16) + C, full f32 precision.
// Args: (neg_a, A, neg_b, B, c_mod, C, reuse_a, reuse_b)
// ---------------------------------------------------------------------------
__device__ __forceinline__ v8f wmma4(v2f a, v2f b, v8f c) {
  return __builtin_amdgcn_wmma_f32_16x16x4_f32(false, a, false, b, (short)0, c,
                                               false, false);
}

// Monotone float<->uint order encoding so atomicMax(uint) == float max.
__device__ __forceinline__ unsigned f2ord(float f) {
  unsigned u = __float_as_uint(f);
  return (u & 0x80000000u) ? ~u : (u | 0x80000000u);
}
__device__ __forceinline__ float ord2f(unsigned u) {
  return __uint_as_float((u & 0x80000000u) ? (u ^ 0x80000000u) : ~u);
}

__device__ __forceinline__ int lowerBound(const int* __restrict__ a, int n, int key) {
  int lo = 0, hi = n;
  while (lo < hi) {
    int mid = (lo + hi) >> 1;
    if (a[mid] < key) lo = mid + 1; else hi = mid;
  }
  return lo;
}

// ---------------------------------------------------------------------------
// Kernel 0: init segment scratch. ord=0 encodes "below every real float".
// ---------------------------------------------------------------------------
__global__ void k_init(unsigned* __restrict__ segmaxOrd, float* __restrict__ segsum) {
  int i = blockIdx.x * blockDim.x + threadIdx.x;
  if (i < N_CRYSTALS * N_HEAD) {
    segmaxOrd[i] = 0u;
    segsum[i] = 0.0f;
  }
}

// ---------------------------------------------------------------------------
// Kernel 1: w[N,32] = fea @ Wk + bk via V_WMMA_F32_16X16X4_F32.
// Block = 256 threads = 8 waves; each wave owns a 16-atom tile (128 atoms/blk).
// Wk is staged k-pair-interleaved: sWkI[(k>>1)*64 + n*2 + (k&1)] so each B
// fragment is one contiguous ds_load_b64 into an even-aligned VGPR pair.
// ---------------------------------------------------------------------------
__global__ void __launch_bounds__(256)
k_logits(const float* __restrict__ fea, const float* __restrict__ Wk,
         const float* __restrict__ bk, const int* __restrict__ owner,
         float* __restrict__ wOut, unsigned* __restrict__ segmaxOrd, int N) {
  __shared__ __align__(16) float sWkI[D_FEA * N_HEAD];  // 8 KB interleaved
  __shared__ float sBk[N_HEAD];

  int tid = threadIdx.x;
  for (int j = tid; j < D_FEA * N_HEAD; j += 256) {
    int k = j >> 5, n = j & 31;  // j = k*32 + n (row-major source)
    sWkI[(k >> 1) * 64 + n * 2 + (k & 1)] = Wk[j];
  }
  if (tid < N_HEAD) sBk[tid] = bk[tid];
  __syncthreads();

  int wave = tid >> 5, lane = tid & 31;
  int l16 = lane & 15, half = (lane >> 4) & 1;
  int rowBase = blockIdx.x * 128 + wave * 16;

  // A-fragment source row for this lane (clamped; tail results discarded).
  int aRow = rowBase + l16;
  if (aRow > N - 1) aRow = N - 1;
  const float* feaRow = fea + (size_t)aRow * D_FEA;

  v8f acc0 = {0, 0, 0, 0, 0, 0, 0, 0};
  v8f acc1 = {0, 0, 0, 0, 0, 0, 0, 0};

#pragma unroll
  for (int k0 = 0; k0 < D_FEA; k0 += 4) {
    int kh = k0 + half * 2;  // lanes 0-15: K=k0,k0+1 ; lanes 16-31: K=k0+2,k0+3
    v2f a = *(const v2f*)(feaRow + kh);            // global_load_b64
    int q = kh >> 1;
    v2f b0 = *(const v2f*)&sWkI[q * 64 + l16 * 2];         // ds_load_b64
    v2f b1 = *(const v2f*)&sWkI[q * 64 + (16 + l16) * 2];  // ds_load_b64
    acc0 = wmma4(a, b0, acc0);
    acc1 = wmma4(a, b1, acc1);
  }

  // C layout: VGPR i -> M = i + 8*half, N = l16 (+16 for acc1).
#pragma unroll
  for (int i = 0; i < 8; ++i) {
    int row = rowBase + i + 8 * half;
    if (row < N) {
      int g = owner[row];
      int n0 = l16, n1 = 16 + l16;
      float w0 = acc0[i] + sBk[n0];
      float w1 = acc1[i] + sBk[n1];
      wOut[(size_t)row * N_HEAD + n0] = w0;
      wOut[(size_t)row * N_HEAD + n1] = w1;
      atomicMax(&segmaxOrd[g * N_HEAD + n0], f2ord(w0));
      atomicMax(&segmaxOrd[g * N_HEAD + n1], f2ord(w1));
    }
  }
}

// ---------------------------------------------------------------------------
// Kernel 2: e = exp(w - segmax[owner]) in place; atomicAdd into segsum.
// ---------------------------------------------------------------------------
__global__ void k_softmax_e(float* __restrict__ w, const int* __restrict__ owner,
                            const unsigned* __restrict__ segmaxOrd,
                            float* __restrict__ segsum, int N) {
  long idx = (long)blockIdx.x * blockDim.x + threadIdx.x;
  if (idx >= (long)N * N_HEAD) return;
  int h = (int)(idx & (N_HEAD - 1));
  int row = (int)(idx >> 5);
  int g = owner[row];
  float mx = ord2f(segmaxOrd[g * N_HEAD + h]);
  float e = __expf(w[idx] - mx);
  w[idx] = e;
  atomicAdd(&segsum[g * N_HEAD + h], e);
}

// ---------------------------------------------------------------------------
// Kernel 3: per-crystal readout: crystal[g] = fea_segᵀ(64 x K) @ p_seg(K x 32)
// One block (128 thr = 4 waves) per crystal; atom_owner sorted -> binary
// search segment bounds. 32-atom chunks staged in LDS with k-pair-interleaved
// layouts; fea chunk is copied with GLOBAL_LOAD_ASYNC_TO_LDS_B32 (per-lane
// scatter, ASYNCcnt-tracked, EXEC-predicated for the tail); zero padding and
// the p chunk go through the normal DS path. 4 waves split K; 4 M-tiles x
// 2 N-tiles of f32 WMMA; LDS reduce of wave partials.
// ---------------------------------------------------------------------------
__global__ void __launch_bounds__(128)
k_readout(const float* __restrict__ fea, const float* __restrict__ e,
          const float* __restrict__ segsum, const int* __restrict__ owner,
          float* __restrict__ out, int N) {
  // Interleaved: element (atom a, col c) lives at [ (a>>1)*2*COLS + c*2 + (a&1) ]
  __shared__ __align__(16) float sFeaI[32 * D_FEA];       // 8 KB
  __shared__ __align__(16) float sPI[32 * N_HEAD];        // 4 KB
  __shared__ float sInv[N_HEAD];
  __shared__ float sAcc[4 * D_FEA * N_HEAD];              // 32 KB wave partials

  int g = blockIdx.x;
  int tid = threadIdx.x;
  int lo = lowerBound(owner, N, g);
  int hiEnd = lowerBound(owner, N, g + 1);

  if (tid < N_HEAD) {
    float s = segsum[g * N_HEAD + tid];
    sInv[tid] = (s > 0.0f) ? 1.0f / s : 0.0f;  // empty crystal -> zeros
  }
  __syncthreads();

  int wave = tid >> 5, lane = tid & 31;
  int l16 = lane & 15, half = (lane >> 4) & 1;

  v8f acc[4][2];
  v8f zero = {0, 0, 0, 0, 0, 0, 0, 0};
#pragma unroll
  for (int mt = 0; mt < 4; ++mt)
#pragma unroll
    for (int nt = 0; nt < 2; ++nt) acc[mt][nt] = zero;

  for (int base = lo; base < hiEnd; base += 32) {
    int cnt = hiEnd - base;
    if (cnt > 32) cnt = 32;

    // Stage fea chunk [32 atoms][64 dims] -> interleaved LDS.
    // Valid lanes: async global->LDS scatter copy; pad lanes: DS zero store.
    for (int jj = tid; jj < 32 * D_FEA; jj += 128) {
      int a = jj >> 6, d = jj & 63;
      int dst = (a >> 1) * 128 + d * 2 + (a & 1);
      if (a < cnt) {
        unsigned ldsOff = (unsigned)(uintptr_t)&sFeaI[dst];
        unsigned gOff = (unsigned)(((unsigned)(base + a) * D_FEA + d) * 4u);
        asm volatile("global_load_async_to_lds_b32 %0, %1, %2"
                     :: "v"(ldsOff), "v"(gOff), "s"(fea)
                     : "memory");
      } else {
        sFeaI[dst] = 0.0f;
      }
    }
    // Stage p chunk [32 atoms][32 heads] = e * 1/segsum -> interleaved LDS.
    for (int jj = tid; jj < 32 * N_HEAD; jj += 128) {
      int a = jj >> 5, h = jj & 31;
      int dst = (a >> 1) * 64 + h * 2 + (a & 1);
      sPI[dst] = (a < cnt) ? e[(size_t)(base + a) * N_HEAD + h] * sInv[h] : 0.0f;
    }
    asm volatile("s_wait_asynccnt 0" ::: "memory");
    __syncthreads();

    // Software prefetch of the next chunk -> global_prefetch_b8.
    if (base + 32 < hiEnd) {
      int pr = base + 32 + lane;
      if (pr < hiEnd) __builtin_prefetch(fea + (size_t)pr * D_FEA, 0, 3);
    }

    // Each wave handles k-steps {2*wave, 2*wave+1} (atoms [8w, 8w+8)).
#pragma unroll
    for (int s = 0; s < 2; ++s) {
      int kk = (wave * 2 + s) * 4 + half * 2;   // even atom index of this pair
      int q = kk >> 1;                          // atom-pair index
      v2f aF[4], bF[2];
#pragma unroll
      for (int mt = 0; mt < 4; ++mt) {
        int m = mt * 16 + l16;  // feature dim (transposed read)
        aF[mt] = *(const v2f*)&sFeaI[q * 128 + m * 2];  // ds_load_b64
      }
#pragma unroll
      for (int nt = 0; nt < 2; ++nt) {
        int n = nt * 16 + l16;
        bF[nt] = *(const v2f*)&sPI[q * 64 + n * 2];     // ds_load_b64
      }
#pragma unroll
      for (int mt = 0; mt < 4; ++mt)
#pragma unroll
        for (int nt = 0; nt < 2; ++nt)
          acc[mt][nt] = wmma4(aF[mt], bF[nt], acc[mt][nt]);
    }
    __syncthreads();  // all waves done reading before restaging LDS
  }

  // Dump wave partials at final [d*32+h] positions, then reduce 4 ways.
#pragma unroll
  for (int mt = 0; mt < 4; ++mt)
#pragma unroll
    for (int nt = 0; nt < 2; ++nt)
#pragma unroll
      for (int i = 0; i < 8; ++i) {
        int d = mt * 16 + i + 8 * half;
        int h = nt * 16 + l16;
        sAcc[wave * (D_FEA * N_HEAD) + d * N_HEAD + h] = acc[mt][nt][i];
      }
  __syncthreads();

  for (int j = tid; j < D_FEA * N_HEAD; j += 128) {
    out[(size_t)g * (D_FEA * N_HEAD) + j] =
        sAcc[j] + sAcc[2048 + j] + sAcc[4096 + j] + sAcc[6144 + j];
  }
}

// ---------------------------------------------------------------------------
extern "C" void kernel_launch(void* const* d_in, const int* in_sizes, int n_in,
                              void* d_out, int out_size, void* d_ws, size_t ws_size,
                              hipStream_t stream) {
  const float* fea  = (const float*)d_in[0];
  const float* Wk   = (const float*)d_in[1];
  const float* bk   = (const float*)d_in[2];
  const int* owner  = (const int*)d_in[4];  // d_in[3] (atomic_numbers) unused
  float* out = (float*)d_out;
  int N = in_sizes[0] / D_FEA;

  // Workspace layout: w/e [N*32] f32 | segmaxOrd [2048*32] u32 | segsum [2048*32] f32
  float* wsW = (float*)d_ws;
  unsigned* segmaxOrd = (unsigned*)(wsW + (size_t)N * N_HEAD);
  float* segsum = (float*)(segmaxOrd + N_CRYSTALS * N_HEAD);

  k_init<<<(N_CRYSTALS * N_HEAD + 255) / 256, 256, 0, stream>>>(segmaxOrd, segsum);
  k_logits<<<(N + 127) / 128, 256, 0, stream>>>(fea, Wk, bk, owner, wsW, segmaxOrd, N);
  long nw = (long)N * N_HEAD;
  k_softmax_e<<<(int)((nw + 255) / 256), 256, 0, stream>>>(wsW, owner, segmaxOrd, segsum, N);
  k_readout<<<N_CRYSTALS, 128, 0, stream>>>(fea, wsW, segsum, owner, out, N);
}